// umbrella_repsurf_43052752175168
// MI455X (gfx1250) — compile-verified
//
#include <hip/hip_runtime.h>

typedef __attribute__((ext_vector_type(2))) float v2f;
typedef __attribute__((ext_vector_type(8))) float v8f;

#define NPTS 4096
#define KNBR 9
#define TOPK 10
#define CHUNK 512

__device__ __forceinline__ void wave_lds_fence() {
  // DS ops from one wave complete in order; stop compiler/scheduler reordering
  // and drain DScnt so cross-lane LDS reads are safe.
  __builtin_amdgcn_wave_barrier();
  asm volatile("s_wait_dscnt 0x0" ::: "memory");
}

// ---------------------------------------------------------------------------
// KNN via V_WMMA_F32_16X16X4_F32 with swapped operands:
//   A = 16 candidate columns (M), B = this wave's 16 rows (N), K = 3 coords.
// C layout then gives each lane 8 candidate columns for ONE row -> top-10
// insertion happens straight out of the accumulator registers, no transpose.
// dist = 2*dot - sq_row - sq_col; self dropped at the final two-half merge.
// ---------------------------------------------------------------------------
__global__ __launch_bounds__(128) void knn_kernel(const float* __restrict__ x,
                                                  unsigned short* __restrict__ nbr) {
  __shared__ float4 cols[CHUNK];               // staged columns x,y,z,(pad) 8 KB
  __shared__ alignas(16) float sqc[CHUNK];     // column squared norms    2 KB
  __shared__ float  mval[4][32][TOPK];
  __shared__ int    midx[4][32][TOPK];

  const int b    = blockIdx.x >> 6;            // NPTS/64 = 64 row-blocks/batch
  const int blk  = blockIdx.x & 63;
  const int tid  = threadIdx.x;
  const int w    = tid >> 5;
  const int lane = tid & 31;
  const int half = lane >> 4;                  // 0: K=0,1 / cols M=0-7 ; 1: K=2,pad / cols M=8-15
  const int l15  = lane & 15;

  const float* xb = x + (size_t)b * NPTS * 3;
  const int row0  = blk * 64 + w * 16;

  // B-matrix (4x16, KxN): N = l15 = our row; lanes<16 carry K=0,1; lanes>=16 K=2,pad
  const int rm = row0 + l15;
  const float rc0 = xb[rm * 3 + 0], rc1 = xb[rm * 3 + 1], rc2 = xb[rm * 3 + 2];
  v2f bmat;
  bmat.x = half ? rc2 : rc0;
  bmat.y = half ? 0.0f : rc1;
  const float sqr = rc0 * rc0 + rc1 * rc1 + rc2 * rc2;   // this lane's row |x|^2

  float bval[TOPK];
  int   bidx[TOPK];
#pragma unroll
  for (int t = 0; t < TOPK; ++t) { bval[t] = -3.4e38f; bidx[t] = 0x7fffffff; }

  const float4* sq4 = (const float4*)sqc;

  for (int j0 = 0; j0 < NPTS; j0 += CHUNK) {
    __syncthreads();                           // protect previous chunk's readers
    for (int i = tid; i < CHUNK; i += 128) {
      const int p = j0 + i;
      const float cx = xb[p * 3 + 0], cy = xb[p * 3 + 1], cz = xb[p * 3 + 2];
      cols[i] = make_float4(cx, cy, cz, 0.0f);
      sqc[i]  = cx * cx + cy * cy + cz * cz;
    }
    __syncthreads();

    for (int j = 0; j < CHUNK; j += 16) {
      // A-matrix (16x4, MxK): M = l15 = candidate column j+l15
      const float4 p = cols[j + l15];
      v2f amat;
      amat.x = half ? p.z : p.x;
      amat.y = half ? 0.0f : p.y;

      // the 8 candidate-column squared norms this lane will score
      const int sbase = (j + 8 * half) >> 2;
      const float4 s0 = sq4[sbase];
      const float4 s1 = sq4[sbase + 1];
      const float sv[8] = {s0.x, s0.y, s0.z, s0.w, s1.x, s1.y, s1.z, s1.w};

      v8f c = {};
      c = __builtin_amdgcn_wmma_f32_16x16x4_f32(false, amat, false, bmat,
                                                (short)0, c, false, false);

      // lane owns (row l15, candidate cols j0+j+8*half+v), straight from regs
#pragma unroll
      for (int v = 0; v < 8; ++v) {
        const float d = 2.0f * c[v] - sqr - sv[v];
        const int  ji = j0 + j + 8 * half + v;
        if (d > bval[TOPK - 1]) {              // strict: earlier index wins ties
          bval[TOPK - 1] = d; bidx[TOPK - 1] = ji;
#pragma unroll
          for (int q = TOPK - 1; q > 0; --q) {
            if (bval[q] > bval[q - 1]) {
              const float tv = bval[q]; bval[q] = bval[q - 1]; bval[q - 1] = tv;
              const int   ti = bidx[q]; bidx[q] = bidx[q - 1]; bidx[q - 1] = ti;
            }
          }
        }
      }
    }
  }

  // merge the two half-lists of each row, skip element 0 (self, dist ~ 0)
#pragma unroll
  for (int t = 0; t < TOPK; ++t) { mval[w][lane][t] = bval[t]; midx[w][lane][t] = bidx[t]; }
  wave_lds_fence();

  if (lane < 16) {
    const int gi = b * NPTS + row0 + lane;
    unsigned short* o = nbr + (size_t)gi * KNBR;
    int pa = 0, pb = 0;
    for (int t = 0; t < TOPK; ++t) {
      const float va = mval[w][lane][pa];
      const float vb = mval[w][lane + 16][pb];
      const int   ia = midx[w][lane][pa];
      const int   ib = midx[w][lane + 16][pb];
      const bool  ta = (va > vb) || (va == vb && ia <= ib);
      const int   ch = ta ? ia : ib;
      if (ta) pa++; else pb++;
      if (t > 0) o[t - 1] = (unsigned short)ch;
    }
  }
}

// ---------------------------------------------------------------------------
// Per-point umbrella features + 2x (7x7 linear, BN, ReLU) + 7x7 linear + max-pool over k.
// ---------------------------------------------------------------------------
__global__ __launch_bounds__(128) void feat_kernel(
    const float* __restrict__ x, const unsigned short* __restrict__ nbr,
    const float* __restrict__ W1, const float* __restrict__ g1,
    const float* __restrict__ b1, const float* __restrict__ m1,
    const float* __restrict__ v1, const float* __restrict__ W2,
    const float* __restrict__ bias2, const float* __restrict__ g2,
    const float* __restrict__ b2, const float* __restrict__ m2,
    const float* __restrict__ v2, const float* __restrict__ W3,
    const float* __restrict__ bias3, float* __restrict__ out, int total) {
  __shared__ float wA[49], wB[49], wC[49];
  __shared__ float sc1[7], sh1[7], sc2[7], sh2[7], b3s[7];

  const int t = threadIdx.x;
  if (t < 49) { wA[t] = W1[t]; wB[t] = W2[t]; wC[t] = W3[t]; }
  if (t < 7) {
    const float s1 = g1[t] * rsqrtf(v1[t] + 1e-5f);
    sc1[t] = s1; sh1[t] = b1[t] - m1[t] * s1;
    const float s2 = g2[t] * rsqrtf(v2[t] + 1e-5f);
    sc2[t] = s2; sh2[t] = (bias2[t] - m2[t]) * s2 + b2[t];
    b3s[t] = bias3[t];
  }
  __syncthreads();

  const int gid = blockIdx.x * 128 + t;
  if (gid >= total) return;
  const int b = gid / NPTS, n = gid % NPTS;
  const float* xb = x + (size_t)b * NPTS * 3;
  const float px = xb[n * 3 + 0], py = xb[n * 3 + 1], pz = xb[n * 3 + 2];

  const unsigned short* nb = nbr + (size_t)gid * KNBR;
  float sx[9], sy[9], sz[9], ph[9];
#pragma unroll
  for (int k = 0; k < 9; ++k) {
    const int id = nb[k];
    sx[k] = xb[id * 3 + 0] - px;
    sy[k] = xb[id * 3 + 1] - py;
    sz[k] = xb[id * 3 + 2] - pz;
    ph[k] = atan2f(sy[k], sx[k]);     // same ordering as phi = atan2/(2pi)+0.5
  }

  // stable bubble sort ascending by phi (matches stable jnp.argsort)
#pragma unroll
  for (int a = 0; a < 8; ++a)
#pragma unroll
    for (int k = 0; k < 8 - a; ++k)
      if (ph[k] > ph[k + 1]) {
        float tv;
        tv = ph[k]; ph[k] = ph[k + 1]; ph[k + 1] = tv;
        tv = sx[k]; sx[k] = sx[k + 1]; sx[k + 1] = tv;
        tv = sy[k]; sy[k] = sy[k + 1]; sy[k + 1] = tv;
        tv = sz[k]; sz[k] = sz[k + 1]; sz[k + 1] = tv;
      }

  // sign mask from x-component of first (k=0) normal
  const float cr0x = sy[0] * sz[1] - sz[0] * sy[1] + 1e-5f;
  const float mask = (cr0x > 0.0f) ? 1.0f : -1.0f;

  float pooled[7];
#pragma unroll
  for (int o = 0; o < 7; ++o) pooled[o] = -3.4e38f;

#pragma unroll
  for (int k = 0; k < 9; ++k) {
    const int k2 = (k + 1) % 9;
    const float ax = sx[k],  ay = sy[k],  az = sz[k];
    const float bx = sx[k2], by = sy[k2], bz = sz[k2];
    const float cex = 0.5f * (ax + bx), cey = 0.5f * (ay + by), cez = 0.5f * (az + bz);
    const float crx = ay * bz - az * by + 1e-5f;
    const float cry = az * bx - ax * bz + 1e-5f;
    const float crz = ax * by - ay * bx + 1e-5f;
    const float inv = mask / sqrtf(crx * crx + cry * cry + crz * crz);
    const float nx = crx * inv, ny = cry * inv, nz = crz * inv;
    const float pos = (cex * nx + cey * ny + cez * nz) * 0.57735026918962576f;
    const float f[7] = {cex, cey, cez, nx, ny, nz, pos};

    float h1[7];
#pragma unroll
    for (int o = 0; o < 7; ++o) {
      float acc = 0.0f;
#pragma unroll
      for (int cc = 0; cc < 7; ++cc) acc += wA[o * 7 + cc] * f[cc];
      h1[o] = fmaxf(acc * sc1[o] + sh1[o], 0.0f);
    }
    float h2[7];
#pragma unroll
    for (int o = 0; o < 7; ++o) {
      float acc = 0.0f;
#pragma unroll
      for (int cc = 0; cc < 7; ++cc) acc += wB[o * 7 + cc] * h1[cc];
      h2[o] = fmaxf(acc * sc2[o] + sh2[o], 0.0f);
    }
#pragma unroll
    for (int o = 0; o < 7; ++o) {
      float acc = b3s[o];
#pragma unroll
      for (int cc = 0; cc < 7; ++cc) acc += wC[o * 7 + cc] * h2[cc];
      pooled[o] = fmaxf(pooled[o], acc);
    }
  }

  float* op = out + (size_t)gid * 10;
  op[0] = px; op[1] = py; op[2] = pz;
#pragma unroll
  for (int o = 0; o < 7; ++o) op[3 + o] = pooled[o];
}

extern "C" void kernel_launch(void* const* d_in, const int* in_sizes, int n_in,
                              void* d_out, int out_size, void* d_ws, size_t ws_size,
                              hipStream_t stream) {
  const float* x     = (const float*)d_in[0];
  const float* W1    = (const float*)d_in[1];
  const float* g1    = (const float*)d_in[2];
  const float* b1    = (const float*)d_in[3];
  const float* m1    = (const float*)d_in[4];
  const float* v1    = (const float*)d_in[5];
  const float* W2    = (const float*)d_in[6];
  const float* bias2 = (const float*)d_in[7];
  const float* g2    = (const float*)d_in[8];
  const float* b2    = (const float*)d_in[9];
  const float* m2    = (const float*)d_in[10];
  const float* v2    = (const float*)d_in[11];
  const float* W3    = (const float*)d_in[12];
  const float* bias3 = (const float*)d_in[13];

  const int B = in_sizes[0] / (NPTS * 3);
  unsigned short* nbr = (unsigned short*)d_ws;   // B*NPTS*9 u16 = 576 KB for B=8

  knn_kernel<<<dim3(B * (NPTS / 64)), dim3(128), 0, stream>>>(x, nbr);

  const int total = B * NPTS;
  feat_kernel<<<dim3((total + 127) / 128), dim3(128), 0, stream>>>(
      x, nbr, W1, g1, b1, m1, v1, W2, bias2, g2, b2, m2, v2, W3, bias3,
      (float*)d_out, total);
}